// Or_14705968021855
// MI455X (gfx1250) — compile-verified
//
#include <hip/hip_runtime.h>

#ifndef __has_builtin
#define __has_builtin(x) 0
#endif

#if __has_builtin(__builtin_amdgcn_global_load_async_to_lds_b128)
#define HAVE_ASYNC_LDS 1
#else
#define HAVE_ASYNC_LDS 0
#endif

#define AS3 __attribute__((address_space(3)))

typedef float f32x4 __attribute__((ext_vector_type(4)));
typedef float f32x2 __attribute__((ext_vector_type(2)));
typedef int   i32x4 __attribute__((ext_vector_type(4)));

static constexpr int kBatch = 128;
static constexpr int kInF   = 2048;
static constexpr int kOutF  = 512;

static constexpr int TB      = 16;   // batch rows per block
static constexpr int TO      = 64;   // out cols per block
static constexpr int KC      = 32;   // k-chunk
static constexpr int THREADS = 256;  // 8 wave32s
static constexpr int NKT     = kInF / KC;

// 16-byte global->LDS copy: async DMA on gfx1250, sync fallback otherwise.
__device__ __forceinline__ void cp16_g2l(const float* __restrict__ src,
                                         float* __restrict__ dst) {
#if HAVE_ASYNC_LDS
  __builtin_amdgcn_global_load_async_to_lds_b128(
      (i32x4*)src, (AS3 i32x4*)dst, /*offset=*/0, /*cpol=*/0);
#else
  *(f32x4*)dst = *(const f32x4*)src;
#endif
}

__device__ __forceinline__ void wait_lds_ready() {
#if HAVE_ASYNC_LDS
#if __has_builtin(__builtin_amdgcn_s_wait_asynccnt)
  __builtin_amdgcn_s_wait_asynccnt(0);
#else
  asm volatile("s_wait_asynccnt 0" ::: "memory");
#endif
#endif
  __syncthreads();
}

// out[b,o] = max_i min(m[b,i], clip(w[i,o],0,1))
// With m in [0,1]: min(m, clip(w,0,1)) == max(0, min(m, w)); acc init 0
// absorbs the max(0,.), so the weight clamp costs zero instructions.
__global__ __launch_bounds__(THREADS) void Or_maxmin_kernel(
    const float* __restrict__ m, const float* __restrict__ w,
    float* __restrict__ out) {
  __shared__ float ms[2][TB][KC];  // 4 KB
  __shared__ float ws[2][KC][TO];  // 16 KB

  const int tid = threadIdx.x;
  const int ob  = blockIdx.x * TO;  // out-col base
  const int bb  = blockIdx.y * TB;  // batch base

  // Stage one k-chunk into LDS buffer `buf` (async, overlapped with compute).
  auto stage = [&](int buf, int kbase) {
    // w tile: KC x TO floats = 512 x f32x4 chunks, 2 per thread (no transpose)
#pragma unroll
    for (int r = 0; r < 2; ++r) {
      const int c   = tid + r * THREADS;  // 0..511
      const int row = c >> 4;             // k within chunk, 0..31
      const int col = (c & 15) << 2;      // o offset, 0..60
      cp16_g2l(w + (size_t)(kbase + row) * kOutF + (ob + col),
               &ws[buf][row][col]);
    }
    // m tile: TB x KC floats = 128 x f32x4 chunks, threads 0..127
    if (tid < (TB * KC) / 4) {
      const int row = tid >> 3;        // b within tile, 0..15
      const int col = (tid & 7) << 2;  // k offset, 0..28
      cp16_g2l(m + (size_t)(bb + row) * kInF + (kbase + col),
               &ms[buf][row][col]);
    }
  };

  // 2x2 register tile per thread: b = bb+tb..tb+1, o = ob+to..to+1
  const int to = (tid & 31) * 2;
  const int tb = (tid >> 5) * 2;

  float acc00 = 0.f, acc01 = 0.f, acc10 = 0.f, acc11 = 0.f;

  stage(0, 0);
  wait_lds_ready();

  for (int kt = 0; kt < NKT; ++kt) {
    const int buf = kt & 1;
    if (kt + 1 < NKT) stage(buf ^ 1, (kt + 1) * KC);  // prefetch next chunk

#pragma unroll
    for (int kk = 0; kk < KC; kk += 4) {
      const f32x4 a0 = *(const f32x4*)&ms[buf][tb][kk];      // broadcast b128
      const f32x4 a1 = *(const f32x4*)&ms[buf][tb + 1][kk];
      const f32x2 w0 = *(const f32x2*)&ws[buf][kk + 0][to];  // b64, no conflicts
      const f32x2 w1 = *(const f32x2*)&ws[buf][kk + 1][to];
      const f32x2 w2 = *(const f32x2*)&ws[buf][kk + 2][to];
      const f32x2 w3 = *(const f32x2*)&ws[buf][kk + 3][to];

      // max(acc, max(t,t)) -> single v_max3_num_f32: 1.5 VALU ops per (out,k)
      acc00 = fmaxf(acc00, fmaxf(fminf(a0.x, w0.x), fminf(a0.y, w1.x)));
      acc00 = fmaxf(acc00, fmaxf(fminf(a0.z, w2.x), fminf(a0.w, w3.x)));

      acc01 = fmaxf(acc01, fmaxf(fminf(a0.x, w0.y), fminf(a0.y, w1.y)));
      acc01 = fmaxf(acc01, fmaxf(fminf(a0.z, w2.y), fminf(a0.w, w3.y)));

      acc10 = fmaxf(acc10, fmaxf(fminf(a1.x, w0.x), fminf(a1.y, w1.x)));
      acc10 = fmaxf(acc10, fmaxf(fminf(a1.z, w2.x), fminf(a1.w, w3.x)));

      acc11 = fmaxf(acc11, fmaxf(fminf(a1.x, w0.y), fminf(a1.y, w1.y)));
      acc11 = fmaxf(acc11, fmaxf(fminf(a1.z, w2.y), fminf(a1.w, w3.y)));
    }
    wait_lds_ready();
  }

  const f32x2 r0 = {acc00, acc01};
  const f32x2 r1 = {acc10, acc11};
  *(f32x2*)(out + (size_t)(bb + tb) * kOutF + ob + to)     = r0;
  *(f32x2*)(out + (size_t)(bb + tb + 1) * kOutF + ob + to) = r1;
}

extern "C" void kernel_launch(void* const* d_in, const int* in_sizes, int n_in,
                              void* d_out, int out_size, void* d_ws,
                              size_t ws_size, hipStream_t stream) {
  const float* m = (const float*)d_in[0];   // [128, 2048]
  const float* w = (const float*)d_in[1];   // [2048, 512]
  float* out     = (float*)d_out;           // [128, 512]

  dim3 grid(kOutF / TO, kBatch / TB);  // (8, 8) = 64 blocks x 8 waves
  Or_maxmin_kernel<<<grid, THREADS, 0, stream>>>(m, w, out);
}